// GNNEncoder_43525198577951
// MI455X (gfx1250) — compile-verified
//
#include <hip/hip_runtime.h>

#define NN_D      64
#define DEPTH_L   4
#define BN_EPS_F  1e-5f
#define STATBLK   400

typedef float v2f __attribute__((ext_vector_type(2)));
typedef float v8f __attribute__((ext_vector_type(8)));

// ---------------- precompute: degree / dinv / edge norm ----------------

__global__ void k_fill_one(float* __restrict__ deg, int n) {
    int i = blockIdx.x * blockDim.x + threadIdx.x;
    if (i < n) deg[i] = 1.0f;              // self-loop contributes 1 to every node
}

__global__ void k_deg_accum(const long long* __restrict__ dst, float* __restrict__ deg, int e) {
    int i = blockIdx.x * blockDim.x + threadIdx.x;
    if (i < e) atomicAdd(&deg[(int)dst[i]], 1.0f);
}

__global__ void k_dinv(float* __restrict__ deg, int n) {
    int i = blockIdx.x * blockDim.x + threadIdx.x;
    if (i < n) deg[i] = rsqrtf(deg[i]);    // deg >= 1 always (self loops)
}

__global__ void k_norm(const long long* __restrict__ src, const long long* __restrict__ dst,
                       const float* __restrict__ dinv, float* __restrict__ norm, int e, int n) {
    long long i = (long long)blockIdx.x * blockDim.x + threadIdx.x;
    long long total = (long long)e + n;
    if (i >= total) return;
    int s, d;
    if (i < e) { s = (int)src[i]; d = (int)dst[i]; } else { s = d = (int)(i - e); }
    norm[i] = dinv[s] * dinv[d];
}

// ---------------- GEMM: out[N,64] = in[N,64] @ W[64,64] via V_WMMA_F32_16X16X4_F32 ----------------
// 256 threads = 8 waves; each wave owns one 16-row tile, sweeps 4 column tiles of 16.
// A (16x4 f32): lanes 0-15 hold K={0,1}, lanes 16-31 hold K={2,3}  -> single float2 load per lane.
// B fragments are PRE-PAIRED in LDS: pair index (kc*64 + col)*2 + khalf holds
// (W[kc*4+khalf*2][col], W[kc*4+khalf*2+1][col]) so each fragment is ONE aligned ds_load_b64
// into an even VGPR pair (no v_mov shuffles), and the 32 lanes of a fragment cover all 64
// LDS banks conflict-free.
// C (16x16 f32): VGPR j -> M=j (lanes 0-15) / M=8+j (lanes 16-31), N=lane&15.

__global__ __launch_bounds__(256) void k_gemm_wmma(const float* __restrict__ in,
                                                   const float* __restrict__ W,
                                                   float* __restrict__ out, int n) {
    __shared__ v2f sW2[NN_D * NN_D / 2];     // 2048 pairs = 16 KB
    // cooperative paired fill: i = kc*128 + khalf*64 + col  (coalesced global reads)
    for (int i = threadIdx.x; i < 2048; i += 256) {
        int kc    = i >> 7;
        int khalf = (i >> 6) & 1;
        int col   = i & 63;
        int kr    = kc * 4 + khalf * 2;
        v2f p;
        p.x = W[kr * NN_D + col];
        p.y = W[(kr + 1) * NN_D + col];
        sW2[(kc * NN_D + col) * 2 + khalf] = p;
    }
    __syncthreads();

    int wave  = threadIdx.x >> 5;
    int lane  = threadIdx.x & 31;
    int tile  = blockIdx.x * 8 + wave;
    int ntile = n >> 4;                      // 100000/16 = 6250, exact
    if (tile >= ntile) return;               // wave-uniform: EXEC stays all-ones for WMMA

    int r0    = tile << 4;
    int mlane = lane & 15;
    int khalf = lane >> 4;                   // 0: K{0,1}, 1: K{2,3}

    // hoist all 16 A fragments (K chunks of 4) for this row tile
    v2f a[16];
    const float* rowp = in + (long long)(r0 + mlane) * NN_D + khalf * 2;
#pragma unroll
    for (int kc = 0; kc < 16; ++kc)
        a[kc] = *(const v2f*)(rowp + kc * 4);

    int mrow = r0 + khalf * 8;
#pragma unroll
    for (int ct = 0; ct < 4; ++ct) {
        int col = ct * 16 + mlane;
        const v2f* bp = sW2 + col * 2 + khalf;   // + kc*128 per chunk
        v8f acc = {};
#pragma unroll
        for (int kc = 0; kc < 16; ++kc) {
            v2f b = bp[kc * NN_D * 2];           // one ds_load_b64, even-pair aligned
            acc = __builtin_amdgcn_wmma_f32_16x16x4_f32(
                false, a[kc], false, b, (short)0, acc, false, false);
        }
#pragma unroll
        for (int j = 0; j < 8; ++j)
            out[(long long)(mrow + j) * NN_D + col] = acc[j];
    }
}

// ---------------- aggregation ----------------

__global__ void k_initbias(float* __restrict__ agg, const float* __restrict__ b, long long total) {
    long long i = (long long)blockIdx.x * blockDim.x + threadIdx.x;
    if (i < total) agg[i] = b[(int)(i & (NN_D - 1))];
}

// one wave per edge: lane loads float2 of the 64-wide source row (coalesced 256B),
// scales by norm, scatter-adds with no-return global f32 atomics (hits L2-resident agg).
__global__ __launch_bounds__(256) void k_message(const float* __restrict__ h,
                                                 const long long* __restrict__ src,
                                                 const long long* __restrict__ dst,
                                                 const float* __restrict__ norm,
                                                 float* __restrict__ agg, int e, int n) {
    long long idx  = (long long)blockIdx.x * 8 + (threadIdx.x >> 5);
    int lane       = threadIdx.x & 31;
    long long total = (long long)e + n;
    if (idx >= total) return;
    int s, d;
    if (idx < e) { s = (int)src[idx]; d = (int)dst[idx]; } else { s = d = (int)(idx - e); }
    float nv = norm[idx];
    v2f v = ((const v2f*)(h + (long long)s * NN_D))[lane];
    float* ap = agg + (long long)d * NN_D + lane * 2;
    atomicAdd(ap,     v.x * nv);
    atomicAdd(ap + 1, v.y * nv);
}

// ---------------- BatchNorm stats (deterministic two-pass) + apply ----------------

__global__ __launch_bounds__(256) void k_stats_partial(const float* __restrict__ agg,
                                                       float* __restrict__ partial, int n) {
    int c = threadIdx.x & 63;
    int g = threadIdx.x >> 6;                 // 4 row-groups per block
    float s = 0.f, ss = 0.f;
    for (int r = blockIdx.x * 4 + g; r < n; r += gridDim.x * 4) {
        float v = agg[(long long)r * NN_D + c];
        s += v; ss += v * v;
    }
    __shared__ float shs[256], shss[256];
    shs[threadIdx.x] = s; shss[threadIdx.x] = ss;
    __syncthreads();
    if (g == 0) {
        for (int gg = 1; gg < 4; ++gg) { s += shs[gg * 64 + c]; ss += shss[gg * 64 + c]; }
        partial[(long long)blockIdx.x * 128 + c]      = s;
        partial[(long long)blockIdx.x * 128 + 64 + c] = ss;
    }
}

__global__ void k_stats_final(const float* __restrict__ partial, float* __restrict__ stats,
                              int nblocks, int n) {
    int c = threadIdx.x;                      // 64 threads
    float s = 0.f, ss = 0.f;
    for (int b = 0; b < nblocks; ++b) {       // fixed order -> deterministic
        s  += partial[(long long)b * 128 + c];
        ss += partial[(long long)b * 128 + 64 + c];
    }
    float mean = s / (float)n;
    float var  = ss / (float)n - mean * mean; // biased variance, matches reference
    stats[c]      = mean;
    stats[64 + c] = rsqrtf(var + BN_EPS_F);
}

__global__ void k_bnrelu(const float* __restrict__ agg, const float* __restrict__ stats,
                         const float* __restrict__ gamma, const float* __restrict__ beta,
                         float* __restrict__ out, long long total) {
    long long i = (long long)blockIdx.x * blockDim.x + threadIdx.x;
    if (i >= total) return;
    int c = (int)(i & (NN_D - 1));
    float v = (agg[i] - stats[c]) * stats[64 + c] * gamma[c] + beta[c];
    out[i] = v > 0.f ? v : 0.f;
}

// ---------------- host orchestration ----------------

extern "C" void kernel_launch(void* const* d_in, const int* in_sizes, int n_in,
                              void* d_out, int out_size, void* d_ws, size_t ws_size,
                              hipStream_t stream) {
    const float*     x      = (const float*)d_in[0];
    const long long* eidx   = (const long long*)d_in[1];   // int64 [2, E]
    const float*     Ws     = (const float*)d_in[2];
    const float*     bs     = (const float*)d_in[3];
    const float*     gammas = (const float*)d_in[4];
    const float*     betas  = (const float*)d_in[5];

    const int n = in_sizes[0] / NN_D;     // 100000
    const int e = in_sizes[1] / 2;        // 1250000
    const long long* src = eidx;
    const long long* dst = eidx + e;

    char* wsb = (char*)d_ws;
    size_t off = 0;
    auto alloc = [&](size_t bytes) -> float* {
        float* p = (float*)(wsb + off);
        off += (bytes + 255) & ~(size_t)255;
        return p;
    };
    const long long total = (long long)n * NN_D;
    float* bufA    = alloc((size_t)total * sizeof(float));
    float* bufB    = alloc((size_t)total * sizeof(float));
    float* deg     = alloc((size_t)n * sizeof(float));           // becomes dinv in place
    float* norm    = alloc((size_t)(e + n) * sizeof(float));
    float* partial = alloc((size_t)STATBLK * 128 * sizeof(float));
    float* stats   = alloc(128 * sizeof(float));
    (void)ws_size; (void)n_in; (void)out_size;

    // ---- precompute edge normalization ----
    k_fill_one <<<(n + 255) / 256, 256, 0, stream>>>(deg, n);
    k_deg_accum<<<(e + 255) / 256, 256, 0, stream>>>(dst, deg, e);
    k_dinv     <<<(n + 255) / 256, 256, 0, stream>>>(deg, n);
    k_norm     <<<(int)(((long long)e + n + 255) / 256), 256, 0, stream>>>(src, dst, deg, norm, e, n);

    // ---- layers ----
    const int gemmGrid = ((n >> 4) + 7) / 8;
    const int msgGrid  = (int)(((long long)e + n + 7) / 8);
    const int ewGrid   = (int)((total + 255) / 256);

    const float* hin = x;
    float* tmp = bufA;    // GEMM result (messages read from it)
    float* agg = bufB;    // scatter-add accumulator

    for (int l = 0; l < DEPTH_L; ++l) {
        k_gemm_wmma    <<<gemmGrid, 256, 0, stream>>>(hin, Ws + l * NN_D * NN_D, tmp, n);
        k_initbias     <<<ewGrid,   256, 0, stream>>>(agg, bs + l * NN_D, total);
        k_message      <<<msgGrid,  256, 0, stream>>>(tmp, src, dst, norm, agg, e, n);
        k_stats_partial<<<STATBLK,  256, 0, stream>>>(agg, partial, n);
        k_stats_final  <<<1,        64,  0, stream>>>(partial, stats, STATBLK, n);
        float* hout = (l == DEPTH_L - 1) ? (float*)d_out : tmp;   // tmp is dead after k_message
        k_bnrelu       <<<ewGrid,   256, 0, stream>>>(agg, stats, gammas + l * NN_D,
                                                      betas + l * NN_D, hout, total);
        hin = hout;
        float* t = agg; agg = tmp; tmp = t;   // ping-pong
    }
}